// core_wise_linear_60473139528502
// MI455X (gfx1250) — compile-verified
//
#include <hip/hip_runtime.h>
#include <stdint.h>

// Problem constants (from the reference)
#define BDIM        4096   // batch
#define OUTC        32     // out channels
#define KBLK        128    // k-rows per staged block (segment-aligned)
#define BLOCKS_PER_WG 8    // 8 * 128 = 1024 k-rows per workgroup
#define KSPLIT      17     // 136 blocks total = 17 * 8
#define WG_BATCH    256    // batch columns per workgroup (8 waves * 32)
#define THREADS     256

typedef __attribute__((ext_vector_type(2))) float v2f;
typedef __attribute__((ext_vector_type(8))) float v8f;
typedef __attribute__((ext_vector_type(4))) int   v4i;

// Only assert device builtins during the device pass; __has_builtin for
// AMDGCN builtins is false in the host (x86) pass by construction.
#if defined(__HIP_DEVICE_COMPILE__) && !__has_builtin(__builtin_amdgcn_wmma_f32_16x16x4_f32)
#error "missing __builtin_amdgcn_wmma_f32_16x16x4_f32 for gfx1250"
#endif

#if __has_builtin(__builtin_amdgcn_global_load_async_to_lds_b128) && \
    __has_builtin(__builtin_amdgcn_s_wait_asynccnt)
#define USE_ASYNC_LDS 1
// Exact parameter types per the compiler diagnostic:
//   param 0: v4i in AS(1) ("int __vector(4) __device__ *")
//   param 1: v4i in AS(3) (LDS)
typedef __attribute__((address_space(1))) v4i* gv4ip;
typedef __attribute__((address_space(3))) v4i* lv4ip;
#else
#define USE_ASYNC_LDS 0
#endif

__device__ __forceinline__ v8f wmma_f32(v2f a, v2f b, v8f c) {
  // 8 args: (neg_a, A, neg_b, B, c_mod, C, reuse_a, reuse_b)
  return __builtin_amdgcn_wmma_f32_16x16x4_f32(false, a, false, b,
                                               (short)0, c, false, false);
}

__global__ void zero_out_kernel(float* __restrict__ p, int n) {
  int i = blockIdx.x * blockDim.x + threadIdx.x;
  if (i < n) p[i] = 0.0f;
}

__global__ __launch_bounds__(THREADS, 2)
void tt_gemm_kernel(const float* __restrict__ first,  // (4, 128, 4096)
                    const float* __restrict__ mid,    // (4, 2, 2048, 4096) == (4, 4096, 4096)
                    const float* __restrict__ last,   // (4, 128, 4096)
                    const float* __restrict__ wgt,    // (17408, 32)
                    float* __restrict__ out)          // (4096, 32)
{
  __shared__ float lw[KBLK * OUTC];   // 16 KB weight stage

  const int tid    = threadIdx.x;
  const int wave   = tid >> 5;
  const int lane   = tid & 31;
  const int lane_m = lane & 15;
  const int khalf  = lane >> 4;       // 0: K={0,1}, 1: K={2,3}
  const int colbase = blockIdx.x * WG_BATCH + wave * 32 + lane_m;

  v8f acc00 = {}, acc01 = {}, acc10 = {}, acc11 = {};

  const int blk0 = blockIdx.y * BLOCKS_PER_WG;

  for (int bi = 0; bi < BLOCKS_PER_WG; ++bi) {
    const int bk = blk0 + bi;   // 128-row block index, 0..135

    __syncthreads();  // WAR: previous block's LDS reads complete
    {
      // Stage this block's 128x32 weight tile into LDS
      const float4* src4 = (const float4*)(wgt + (size_t)bk * (KBLK * OUTC));
      float4* dst4 = (float4*)lw;
#if USE_ASYNC_LDS
      for (int i = tid; i < (KBLK * OUTC) / 4; i += THREADS) {
        __builtin_amdgcn_global_load_async_to_lds_b128(
            (gv4ip)(src4 + i), (lv4ip)(dst4 + i), 0, 0);
      }
      __builtin_amdgcn_s_wait_asynccnt(0);
#else
      for (int i = tid; i < (KBLK * OUTC) / 4; i += THREADS) dst4[i] = src4[i];
#endif
    }
    __syncthreads();

    // Resolve which source tensor this 128-row block lives in (uniform, SALU).
    // Per object (4352 rows = 34 blocks): [first:1 blk | mid:32 blks | last:1 blk]
    const int c  = bk / 34;
    const int rb = bk - c * 34;
    const float* seg;
    if (rb == 0)       seg = first + (size_t)c * 128 * BDIM;
    else if (rb <= 32) seg = mid  + ((size_t)c * 4096 + (size_t)(rb - 1) * 128) * BDIM;
    else               seg = last + (size_t)c * 128 * BDIM;

    const float* aptr = seg + (size_t)(2 * khalf) * BDIM + colbase;

#pragma unroll 4
    for (int ks = 0; ks < KBLK / 4; ++ks) {
      const float* p = aptr + (size_t)(4 * ks) * BDIM;

      // Prefetch A stream 4 k-steps (256 KB) ahead: 16 KB row stride
      __builtin_prefetch(p + (size_t)16 * BDIM, 0, 0);
      __builtin_prefetch(p + (size_t)17 * BDIM, 0, 0);

      // A operand: I[k, b] — lane holds rows K=2*khalf+{0,1}, col lane_m (+16)
      float a00 = p[0];            // M-tile 0, K vgpr 0
      float a01 = p[16];           // M-tile 1, K vgpr 0
      float a10 = p[BDIM];         // M-tile 0, K vgpr 1
      float a11 = p[BDIM + 16];    // M-tile 1, K vgpr 1

      // B operand: W[k, o] from LDS — lane holds K=2*khalf+{0,1}, col lane_m (+16)
      const int kk = ks * 4 + 2 * khalf;
      float b00 = lw[kk * OUTC + lane_m];
      float b10 = lw[kk * OUTC + lane_m + 16];
      float b01 = lw[(kk + 1) * OUTC + lane_m];
      float b11 = lw[(kk + 1) * OUTC + lane_m + 16];

      v2f A0 = {a00, a10}, A1 = {a01, a11};
      v2f B0 = {b00, b01}, B1 = {b10, b11};

      acc00 = wmma_f32(A0, B0, acc00);
      acc01 = wmma_f32(A0, B1, acc01);
      acc10 = wmma_f32(A1, B0, acc10);
      acc11 = wmma_f32(A1, B1, acc11);
    }
  }

  // Split-K epilogue: fp32 atomics into out (D layout: vgpr r -> M = r + 8*khalf,
  // lanes 0-15 -> N = lane_m, N-tile 1 -> +16). Rows of `out` are 32 floats, so
  // each half-wave's 16 lanes write 64B contiguous.
  const int mrow = blockIdx.x * WG_BATCH + wave * 32 + 8 * khalf;
#pragma unroll
  for (int r = 0; r < 8; ++r) {
    unsafeAtomicAdd(&out[(size_t)(mrow + r)      * OUTC + lane_m     ], acc00[r]);
    unsafeAtomicAdd(&out[(size_t)(mrow + r)      * OUTC + lane_m + 16], acc01[r]);
    unsafeAtomicAdd(&out[(size_t)(mrow + r + 16) * OUTC + lane_m     ], acc10[r]);
    unsafeAtomicAdd(&out[(size_t)(mrow + r + 16) * OUTC + lane_m + 16], acc11[r]);
  }
}

extern "C" void kernel_launch(void* const* d_in, const int* in_sizes, int n_in,
                              void* d_out, int out_size, void* d_ws, size_t ws_size,
                              hipStream_t stream) {
  const float* first = (const float*)d_in[0];  // (4,128,4096)
  const float* mid   = (const float*)d_in[1];  // (4,2,2048,4096)
  const float* last  = (const float*)d_in[2];  // (4,128,4096)
  const float* wgt   = (const float*)d_in[3];  // (17408,32)
  float* out = (float*)d_out;                  // (4096,32) fp32

  // Zero output (split-K accumulates with atomics); same stream => ordered.
  zero_out_kernel<<<(out_size + THREADS - 1) / THREADS, THREADS, 0, stream>>>(out, out_size);

  dim3 grid(BDIM / WG_BATCH, KSPLIT);  // (16, 17) = 272 workgroups
  tt_gemm_kernel<<<grid, THREADS, 0, stream>>>(first, mid, last, wgt, out);
}